// DecTag_LightGCN_33741263077985
// MI455X (gfx1250) — compile-verified
//
#include <hip/hip_runtime.h>

typedef float v2f __attribute__((ext_vector_type(2)));
typedef float v8f __attribute__((ext_vector_type(8)));

// ---- CDNA5 async global->LDS helpers (ASYNCcnt-tracked DMA, bypasses VGPRs)
__device__ __forceinline__ void async_b128(const float* g, float* l) {
    // LDS byte offset of an addrspace(3) pointer
    unsigned loff = (unsigned)(size_t)(__attribute__((address_space(3))) float*)l;
    asm volatile("global_load_async_to_lds_b128 %0, %1, off"
                 :: "v"(loff), "v"((unsigned long long)(size_t)g)
                 : "memory");
}
__device__ __forceinline__ void wait_async0() {
    asm volatile("s_wait_asynccnt 0" ::: "memory");
}

// ---------------- Fused GEMM + bias + ReLU using V_WMMA_F32_16X16X4_F32 ----
// Block: 256 threads = 8 waves. Block tile: 128 (rows) x 64 (cols).
// Wave tile: 32x32 = 2x2 WMMA 16x16 tiles. K double-buffered through LDS,
// 32 at a time, staged with global_load_async_to_lds_b128.
__global__ __launch_bounds__(256) void gemm_relu_wmma(
    const float* __restrict__ A, const float* __restrict__ B,
    const float* __restrict__ bias, float* __restrict__ C,
    int M, int K, int Nn)
{
    __shared__ float As[2][128 * 36];  // pad 32->36: 16B-aligned, conflict-free
    __shared__ float Bs[2][32 * 68];   // pad 64->68

    const int tid   = threadIdx.x;
    const int lane  = tid & 31;
    const int wave  = tid >> 5;
    const int lrow  = lane & 15;        // M (A) / N (B,C) index within tile
    const int hi    = lane >> 4;        // lane-half
    const int khalf = hi << 1;          // K offset from lane-half
    const int wr    = (wave & 3) * 32;  // wave row offset in block tile
    const int wc    = (wave >> 2) * 32; // wave col offset in block tile
    const int m0    = blockIdx.x * 128;
    const int n0    = blockIdx.y * 64;

    // staging coordinates (fixed per thread)
    const int ac  = tid & 7;            // A: float4 column chunk (0..7)
    const int ar0 = tid >> 3;           // A: row 0..31 (stride 32, 4 chunks)
    const float4 zero4 = make_float4(0.f, 0.f, 0.f, 0.f);

    v8f acc[2][2];
#pragma unroll
    for (int i = 0; i < 2; ++i)
#pragma unroll
        for (int j = 0; j < 2; ++j)
#pragma unroll
            for (int v = 0; v < 8; ++v) acc[i][j][v] = 0.0f;

    // ---- stage one 32-wide K slab into buffer `buf` (async DMA to LDS) ----
    auto stage = [&](int buf, int k0) {
#pragma unroll
        for (int q = 0; q < 4; ++q) {           // A tile: 128x32
            const int r  = ar0 + q * 32;
            const int gm = m0 + r;
            float* ldst = &As[buf][r * 36 + ac * 4];
            if (gm < M) async_b128(A + (size_t)gm * K + k0 + ac * 4, ldst);
            else        *(float4*)ldst = zero4;
        }
#pragma unroll
        for (int q = 0; q < 2; ++q) {           // B tile: 32x64
            const int ch = tid + q * 256;
            const int rr = ch >> 4;
            const int cc = (ch & 15) * 4;
            const int gc = n0 + cc;
            float* ldst = &Bs[buf][rr * 68 + cc];
            if (gc < Nn) async_b128(B + (size_t)(k0 + rr) * Nn + gc, ldst);
            else         *(float4*)ldst = zero4;
        }
    };

    // prologue: fill buffer 0
    stage(0, 0);
    wait_async0();
    __syncthreads();

    for (int k0 = 0; k0 < K; k0 += 32) {
        const int cur = (k0 >> 5) & 1;
        if (k0 + 32 < K) stage(cur ^ 1, k0 + 32);   // overlap DMA with compute

#pragma unroll
        for (int kk = 0; kk < 32; kk += 4) {
            v2f a[2], b[2];
#pragma unroll
            for (int i = 0; i < 2; ++i) {
                // A frag: lane holds A[M=lrow][K = kk+khalf .. +1] (8B aligned)
                const float2 t =
                    *(const float2*)(&As[cur][(wr + i * 16 + lrow) * 36 + kk + khalf]);
                a[i].x = t.x; a[i].y = t.y;
            }
#pragma unroll
            for (int j = 0; j < 2; ++j) {
                // B frag: lane holds B[K = kk+khalf+v][N = lrow]
                b[j].x = Bs[cur][(kk + khalf + 0) * 68 + wc + j * 16 + lrow];
                b[j].y = Bs[cur][(kk + khalf + 1) * 68 + wc + j * 16 + lrow];
            }
#pragma unroll
            for (int i = 0; i < 2; ++i)
#pragma unroll
                for (int j = 0; j < 2; ++j)
                    acc[i][j] = __builtin_amdgcn_wmma_f32_16x16x4_f32(
                        false, a[i], false, b[j], (short)0, acc[i][j], false, false);
        }

        wait_async0();       // next slab's DMA complete (this wave)
        __syncthreads();     // whole workgroup: safe to flip buffers
    }

    // ---- epilogue: bias + ReLU, C/D layout: VGPR v -> M = v + hi*8 ----
#pragma unroll
    for (int i = 0; i < 2; ++i)
#pragma unroll
        for (int j = 0; j < 2; ++j) {
            const int col = n0 + wc + j * 16 + lrow;
            if (col < Nn) {
                const float bv = bias[col];
#pragma unroll
                for (int v = 0; v < 8; ++v) {
                    const int row = m0 + wr + i * 16 + hi * 8 + v;
                    if (row < M) {
                        const float o = acc[i][j][v] + bv;
                        C[(size_t)row * Nn + col] = fmaxf(o, 0.0f);
                    }
                }
            }
        }
}

// ---------------- COO SpMM layer: dst[row] += val * src[col], D=64 ---------
// One wave processes 32 edges: coalesced index loads (+prefetch 8192 edges
// ahead: the 48MB/layer index stream is the only HBM-bound part — the
// 30.7MB embedding table lives in the 192MB L2), shuffle-broadcast, full-wave
// float2 row gather (contiguous 256B), hardware fp32 atomics for the scatter.
__global__ __launch_bounds__(256) void spmm_layer(
    const float* __restrict__ vals, const int* __restrict__ rows,
    const int* __restrict__ cols, const float* __restrict__ src,
    float* __restrict__ dst, int nnz)
{
    const int gwave = (blockIdx.x * blockDim.x + threadIdx.x) >> 5;
    const int lane  = threadIdx.x & 31;
    const long long base = (long long)gwave * 32;
    if (base >= nnz) return;

    float v = 0.0f; int r = 0, c = 0;
    if (base + lane < nnz) {
        v = vals[base + lane];
        r = rows[base + lane];
        c = cols[base + lane];
    }
    // prefetch the index stream far ahead (global_prefetch_b8)
    if (base + 8192 + lane < nnz) {
        __builtin_prefetch(vals + base + 8192 + lane, 0, 1);
        __builtin_prefetch(rows + base + 8192 + lane, 0, 1);
        __builtin_prefetch(cols + base + 8192 + lane, 0, 1);
    }
#pragma unroll 4
    for (int i = 0; i < 32; ++i) {
        if (base + i >= nnz) break;
        const float vi = __shfl(v, i, 32);
        const int   ri = __shfl(r, i, 32);
        const int   ci = __shfl(c, i, 32);
        const float2 x = ((const float2*)(src + (size_t)ci * 64))[lane];
        float* op = dst + (size_t)ri * 64 + lane * 2;
        __hip_atomic_fetch_add(op + 0, vi * x.x, __ATOMIC_RELAXED, __HIP_MEMORY_SCOPE_AGENT);
        __hip_atomic_fetch_add(op + 1, vi * x.y, __ATOMIC_RELAXED, __HIP_MEMORY_SCOPE_AGENT);
    }
}

// ---------------- small elementwise kernels --------------------------------
__global__ void copy_vec(const float* __restrict__ src, float* __restrict__ dst, int n) {
    int i = blockIdx.x * blockDim.x + threadIdx.x;
    if (i < n) dst[i] = src[i];
}
__global__ void init_acc(const float* __restrict__ emb, float* __restrict__ acc,
                         float* __restrict__ z, int n) {
    int i = blockIdx.x * blockDim.x + threadIdx.x;
    if (i < n) { acc[i] = emb[i]; z[i] = 0.0f; }
}
__global__ void accum_zero(float* __restrict__ acc, const float* __restrict__ src,
                           float* __restrict__ z, int n) {
    int i = blockIdx.x * blockDim.x + threadIdx.x;
    if (i < n) { acc[i] += src[i]; z[i] = 0.0f; }
}
__global__ void finalize(const float* __restrict__ acc, float* __restrict__ out, int n) {
    int i = blockIdx.x * blockDim.x + threadIdx.x;
    if (i < n) out[i] = acc[i] * 0.25f;
}

// ---------------- host driver ----------------------------------------------
extern "C" void kernel_launch(void* const* d_in, const int* in_sizes, int n_in,
                              void* d_out, int out_size, void* d_ws, size_t ws_size,
                              hipStream_t stream) {
    (void)in_sizes; (void)n_in; (void)out_size; (void)ws_size;

    const float* X   = (const float*)d_in[0];  // 100000 x 1024
    const float* W1  = (const float*)d_in[1];  // 1024 x 544
    const float* b1  = (const float*)d_in[2];  // 544
    const float* W2  = (const float*)d_in[3];  // 544 x 64
    const float* b2  = (const float*)d_in[4];  // 64
    const float* tag = (const float*)d_in[5];  // 20000 x 64
    const float* gv  = (const float*)d_in[6];  // 4M
    const int*   gr  = (const int*)d_in[7];    // 4M
    const int*   gc  = (const int*)d_in[8];    // 4M
    float* out = (float*)d_out;                // 120000 x 64 (items then tags)

    const int M = 100000, K1n = 1024, N1n = 544, D = 64;
    const int NT = 120000, NNZ = 4000000;
    const int NE = NT * D;                     // 7,680,000

    char* ws = (char*)d_ws;
    float* H   = (float*)ws;                          // 100000*544 f = 217.6 MB
    float* emb = (float*)(ws + 217600000LL);          // 30.72 MB, after H
    // H is dead after GEMM2 -> reuse its region for the SpMM ping-pong + acc:
    float* nA  = (float*)ws;
    float* nB  = (float*)(ws + 30720000LL);
    float* acc = (float*)(ws + 61440000LL);

    // MLP: H = relu(X@W1+b1); emb[:items] = relu(H@W2+b2)
    gemm_relu_wmma<<<dim3(782, 9), 256, 0, stream>>>(X, W1, b1, H, M, K1n, N1n);
    gemm_relu_wmma<<<dim3(782, 1), 256, 0, stream>>>(H, W2, b2, emb, M, N1n, D);
    // emb[items:] = tag_emb
    copy_vec<<<5000, 256, 0, stream>>>(tag, emb + (size_t)M * D, 20000 * D);
    // acc = emb ; nA = 0
    init_acc<<<30000, 256, 0, stream>>>(emb, acc, nA, NE);

    // 3 LightGCN propagation layers
    spmm_layer<<<15625, 256, 0, stream>>>(gv, gr, gc, emb, nA, NNZ);
    accum_zero<<<30000, 256, 0, stream>>>(acc, nA, nB, NE);
    spmm_layer<<<15625, 256, 0, stream>>>(gv, gr, gc, nA, nB, NNZ);
    accum_zero<<<30000, 256, 0, stream>>>(acc, nB, nA, NE);
    spmm_layer<<<15625, 256, 0, stream>>>(gv, gr, gc, nB, nA, NNZ);
    accum_zero<<<30000, 256, 0, stream>>>(acc, nA, nB, NE);  // zeroing nB harmless

    // out = acc / 4
    finalize<<<30000, 256, 0, stream>>>(acc, out, NE);
}